// SplineGCN_15556371546869
// MI455X (gfx1250) — compile-verified
//
#include <hip/hip_runtime.h>

// SplineGCN on MI455X (gfx1250, wave32).
// Pipeline: zero -> WMMA f32 GEMM (Z = feat x all 64 weight mats, stored f16)
//           -> per-edge spline gather/weighted-sum + atomic scatter -> mean+bias.
//
// Z is stored as f16 (204.8 MB ~= L2 size) so the 0.8 GB of per-edge gathers
// resolve mostly in the 192 MB L2 instead of HBM. All accumulation is fp32.

typedef float v2f __attribute__((ext_vector_type(2)));
typedef float v8f __attribute__((ext_vector_type(8)));

#define NNODES 50000
#define NEDGES 1600000
#define IN_F   32
#define OUT_F  32
#define KS     4
#define KMAT   64                 // KS^3
#define ZCOLS  (KMAT * OUT_F)     // 2048

// ---------------------------------------------------------------- kernel 0
__global__ void __launch_bounds__(256)
spline_zero_kernel(float* __restrict__ out, float* __restrict__ deg) {
    int i = blockIdx.x * 256 + threadIdx.x;
    if (i < NNODES * OUT_F) out[i] = 0.0f;
    if (i < NNODES)         deg[i] = 0.0f;
}

// ---------------------------------------------------------------- kernel 1
// Z[n, k*32+o] = sum_i feat[n,i] * W[k,i,o], accumulated fp32 in WMMA, stored f16.
// One wave32 computes a 16x16 tile of Z using 8x V_WMMA_F32_16X16X4_F32.
// A-frag (16x4 f32): lanes 0-15 hold row m, VGPR0=K0,VGPR1=K1; lanes 16-31 K2,K3.
// B-frag (4x16 f32): mirrored (N across lanes, K split across lane halves).
__global__ void __launch_bounds__(256)
spline_gemm_kernel(const float* __restrict__ feat,
                   const float* __restrict__ weight,
                   _Float16* __restrict__ Z) {
    const int lane  = threadIdx.x & 31;
    const int wave  = blockIdx.x * (blockDim.x >> 5) + (threadIdx.x >> 5);
    const int ntile = wave & (ZCOLS / 16 - 1);   // 128 column tiles
    const int mtile = wave >> 7;                 // 3125 row tiles
    const int m0 = mtile * 16, n0 = ntile * 16;
    const int half = lane >> 4;                  // 0: K+{0,1}, 1: K+{2,3}
    const int l16  = lane & 15;

    const int col   = n0 + l16;                                 // Z column
    const int wbase = (col >> 5) * (IN_F * OUT_F) + (col & 31); // weight[k, *, o]

    v8f acc = {};
#pragma unroll
    for (int k = 0; k < IN_F; k += 4) {
        const int kk = k + 2 * half;
        v2f a, b;
        a.x = feat[(m0 + l16) * IN_F + kk];
        a.y = feat[(m0 + l16) * IN_F + kk + 1];
        b.x = weight[wbase + kk * OUT_F];
        b.y = weight[wbase + (kk + 1) * OUT_F];
        acc = __builtin_amdgcn_wmma_f32_16x16x4_f32(
            /*neg_a=*/false, a, /*neg_b=*/false, b,
            /*c_mod=*/(short)0, acc, /*reuse_a=*/false, /*reuse_b=*/false);
    }
    // C/D layout: VGPR r on lane L -> element (M = r + 8*(L>=16), N = L&15)
#pragma unroll
    for (int r = 0; r < 8; ++r)
        Z[(m0 + r + 8 * half) * ZCOLS + n0 + l16] = (_Float16)acc[r];
}

// ---------------------------------------------------------------- kernel 2
// One wave per edge. Lane o handles output feature o.
// Each (edge, s) gather is 32 lanes x 2B = 64B, coalesced; the s and s^1
// corners are adjacent Z rows, so full 128B lines are consumed.
__global__ void __launch_bounds__(256)
spline_edge_kernel(const float* __restrict__ pseudo,
                   const int*   __restrict__ eidx,
                   const _Float16* __restrict__ Z,
                   float* __restrict__ acc_out,
                   float* __restrict__ deg) {
    const int lane = threadIdx.x & 31;
    const int e = blockIdx.x * (blockDim.x >> 5) + (threadIdx.x >> 5);
    if (e >= NEDGES) return;

    const int row = eidx[e];            // target
    const int col = eidx[NEDGES + e];   // source

    // Degree-1 open-spline basis (computed redundantly per lane; the 3 loads
    // are wave-uniform so they coalesce into a single request).
    float frac[3];
    int   bot[3];
#pragma unroll
    for (int d = 0; d < 3; ++d) {
        float v = pseudo[e * 3 + d] * (float)(KS - 1);
        float f = floorf(v);
        f = fminf(fmaxf(f, 0.0f), (float)(KS - 2));
        bot[d]  = (int)f;
        frac[d] = v - f;
    }

    const _Float16* zrow = Z + col * ZCOLS + lane;
    float m = 0.0f;
#pragma unroll
    for (int s = 0; s < 8; ++s) {
        float w = 1.0f;
        int flat = 0, str = 1;
#pragma unroll
        for (int d = 0; d < 3; ++d) {
            const int bit = (s >> d) & 1;
            w *= bit ? frac[d] : (1.0f - frac[d]);
            flat += (bot[d] + bit) * str;
            str *= KS;
        }
        m = fmaf(w, (float)zrow[flat * OUT_F], m);
    }

    atomicAdd(&acc_out[row * OUT_F + lane], m);   // lands in L2 (6.4MB region)
    if (lane == 0) atomicAdd(&deg[row], 1.0f);
}

// ---------------------------------------------------------------- kernel 3
__global__ void __launch_bounds__(256)
spline_finalize_kernel(float* __restrict__ out,
                       const float* __restrict__ deg,
                       const float* __restrict__ bias) {
    int i = blockIdx.x * 256 + threadIdx.x;
    if (i >= NNODES * OUT_F) return;
    float d = fmaxf(deg[i >> 5], 1.0f);
    out[i] = out[i] / d + bias[i & 31];
}

// ---------------------------------------------------------------- launch
extern "C" void kernel_launch(void* const* d_in, const int* in_sizes, int n_in,
                              void* d_out, int out_size, void* d_ws, size_t ws_size,
                              hipStream_t stream) {
    const float* feat   = (const float*)d_in[0];   // [N,32]
    const float* pseudo = (const float*)d_in[1];   // [E,3]
    const float* weight = (const float*)d_in[2];   // [64,32,32]
    const float* bias   = (const float*)d_in[3];   // [32]
    const int*   eidx   = (const int*)  d_in[4];   // [2,E]
    float*       out    = (float*)d_out;           // [N,32]

    // Workspace: Z [N,2048] f16 (204.8 MB) then deg [N] f32.
    _Float16* Z   = (_Float16*)d_ws;
    float*    deg = (float*)((char*)d_ws + (size_t)NNODES * ZCOLS * sizeof(_Float16));

    // 0) zero accumulators (d_out is poisoned by the harness)
    spline_zero_kernel<<<(NNODES * OUT_F + 255) / 256, 256, 0, stream>>>(out, deg);

    // 1) Z = feat x W : 3125 x 128 = 400000 16x16 tiles, 8 waves/block
    spline_gemm_kernel<<<(3125 * 128) / 8, 256, 0, stream>>>(feat, weight, Z);

    // 2) per-edge gather + scatter: 8 edges per 256-thread block
    spline_edge_kernel<<<(NEDGES + 7) / 8, 256, 0, stream>>>(pseudo, eidx, Z, out, deg);

    // 3) mean + bias
    spline_finalize_kernel<<<(NNODES * OUT_F + 255) / 256, 256, 0, stream>>>(out, deg, bias);
}